// DuelForm_WSVectorQuantizer_79894981640746
// MI455X (gfx1250) — compile-verified
//
#include <hip/hip_runtime.h>
#include <math.h>

typedef __attribute__((ext_vector_type(2))) float v2f;
typedef __attribute__((ext_vector_type(8))) float v8f;

#define N_TOK   8192     // H*W*B
#define CDIM    256
#define KCB     1024
#define ENS     64
#define INV_EPS 10.0f    // 1/0.1
#define TEMP_C  0.5f
#define BETA_C  0.25f

// ---- workspace layout (float offsets) ----
#define WS_ZF    0                       // 8192*256
#define WS_COST  (WS_ZF   + N_TOK*CDIM)  // 8192*1024
#define WS_ZN    (WS_COST + N_TOK*KCB)   // 8192
#define WS_CK    (WS_ZN   + N_TOK)       // 1024
#define WS_PHI1  (WS_CK   + KCB)         // 64*1024
#define WS_PHI2  (WS_PHI1 + ENS*KCB)     // 8192
#define WS_HIST  (WS_PHI2 + N_TOK)       // 1024
#define WS_IDX   (WS_HIST + KCB)         // 8192 (int)
#define WS_ACC   (WS_IDX  + N_TOK)       // 8

__global__ __launch_bounds__(256) void vq_init(float* hist, float* acc) {
    int t = threadIdx.x;
    for (int i = t; i < KCB; i += 256) hist[i] = 0.0f;
    if (t < 8) acc[t] = 0.0f;
}

// zf[n*256+c] from z[(b*256+c)*64+hw], n = hw*128+b ; also row norms
__global__ __launch_bounds__(256) void vq_build_zf(const float* __restrict__ z,
                                                   float* __restrict__ zf,
                                                   float* __restrict__ zn) {
    int n = blockIdx.x, t = threadIdx.x;
    int hw = n >> 7, b = n & 127;
    float v = z[(b * CDIM + t) * 64 + hw];
    zf[n * CDIM + t] = v;
    __shared__ float red[256];
    red[t] = v * v; __syncthreads();
    for (int s = 128; s > 0; s >>= 1) { if (t < s) red[t] += red[t + s]; __syncthreads(); }
    if (t == 0) zn[n] = red[0];
}

// codebook row norms + phi1[e,k] = dot(cb[k,:], kan1_w[e,:]) + kan1_b[e]
__global__ __launch_bounds__(256) void vq_phi1_ck(const float* __restrict__ cb,
                                                  const float* __restrict__ w1,
                                                  const float* __restrict__ b1,
                                                  float* __restrict__ phi1,
                                                  float* __restrict__ ck) {
    int k = blockIdx.x, t = threadIdx.x;
    __shared__ float row[256];
    __shared__ float red[256];
    float v = cb[k * CDIM + t];
    row[t] = v; red[t] = v * v; __syncthreads();
    for (int s = 128; s > 0; s >>= 1) { if (t < s) red[t] += red[t + s]; __syncthreads(); }
    if (t == 0) ck[k] = red[0];
    if (t < ENS) {
        const float* w = w1 + t * CDIM;
        float s = 0.0f;
        for (int c = 0; c < CDIM; ++c) s = fmaf(row[c], w[c], s);
        phi1[t * KCB + k] = s + b1[t];
    }
}

// phi2[n] = dot(zf[n,:], kan2_w[e,:]) + kan2_b[e], e = n>>7
__global__ __launch_bounds__(256) void vq_phi2(const float* __restrict__ zf,
                                               const float* __restrict__ w2,
                                               const float* __restrict__ b2,
                                               float* __restrict__ phi2) {
    int n = blockIdx.x, t = threadIdx.x, e = n >> 7;
    __shared__ float red[256];
    red[t] = zf[n * CDIM + t] * w2[e * CDIM + t]; __syncthreads();
    for (int s = 128; s > 0; s >>= 1) { if (t < s) red[t] += red[t + s]; __syncthreads(); }
    if (t == 0) phi2[n] = red[0] + b2[e];
}

// cost[n,k] = zn[n] + ck[k] - 2*dot  via V_WMMA_F32_16X16X4_F32
// WG = 8 waves; wave tile 16x16; WG tile 64 rows x 32 cols
__global__ __launch_bounds__(256) void vq_cost_wmma(const float* __restrict__ zf,
                                                    const float* __restrict__ cb,
                                                    const float* __restrict__ zn,
                                                    const float* __restrict__ ck,
                                                    float* __restrict__ cost) {
    int wave = threadIdx.x >> 5;
    int lane = threadIdx.x & 31;
    int l16 = lane & 15;
    int kh = lane >> 4;                      // K-half select (0: K0/K1, 1: K2/K3)
    int row0 = blockIdx.x * 64 + (wave & 3) * 16;
    int col0 = blockIdx.y * 32 + (wave >> 2) * 16;

    const float* arow = zf + (row0 + l16) * CDIM + 2 * kh;  // A[M=l16][K=2*kh + {0,1}]
    const float* brow = cb + (col0 + l16) * CDIM + 2 * kh;  // B[K][N=l16] = cbT
    v8f acc = {};
    for (int c = 0; c < CDIM; c += 4) {
        v2f a, b;
        a.x = arow[c]; a.y = arow[c + 1];
        b.x = brow[c]; b.y = brow[c + 1];
        acc = __builtin_amdgcn_wmma_f32_16x16x4_f32(false, a, false, b,
                                                    (short)0, acc, false, false);
    }
    int col = col0 + l16;
    float ckv = ck[col];
    for (int g = 0; g < 8; ++g) {
        int row = row0 + g + 8 * kh;         // D layout: M = g + 8*(lane/16), N = lane%16
        cost[row * KCB + col] = zn[row] + ckv - 2.0f * acc[g];
    }
}

// per-row argmin (first-min tiebreak) + usage histogram
__global__ __launch_bounds__(256) void vq_argmin(const float* __restrict__ cost,
                                                 int* __restrict__ idx,
                                                 float* __restrict__ hist) {
    int n = blockIdx.x, t = threadIdx.x;
    const float* row = cost + n * KCB;
    float bv = 3.4e38f; int bi = 0;
    for (int j = 0; j < 4; ++j) {
        int k = t + 256 * j;
        float v = row[k];
        if (v < bv) { bv = v; bi = k; }
    }
    __shared__ float sv[256]; __shared__ int si[256];
    sv[t] = bv; si[t] = bi; __syncthreads();
    for (int s = 128; s > 0; s >>= 1) {
        if (t < s) {
            float vo = sv[t + s]; int io = si[t + s];
            if (vo < sv[t] || (vo == sv[t] && io < si[t])) { sv[t] = vo; si[t] = io; }
        }
        __syncthreads();
    }
    if (t == 0) { idx[n] = si[0]; atomicAdd(&hist[si[0]], 1.0f); }
}

// zq_out[b][c][hw] = codebook[idx[hw*128+b]][c]; tid enumerates output linearly
__global__ __launch_bounds__(256) void vq_gather(const float* __restrict__ cb,
                                                 const int* __restrict__ idx,
                                                 float* __restrict__ out) {
    int tid = blockIdx.x * 256 + threadIdx.x;   // 2,097,152
    int hw = tid & 63;
    int bc = tid >> 6;
    int c  = bc & 255;
    int b  = bc >> 8;
    int n  = hw * 128 + b;
    out[tid] = cb[idx[n] * CDIM + c];
}

// loss1: per-row logsumexp over K of (phi1[e,k]-cost[n,k])/eps ; atomic sum of lse
__global__ __launch_bounds__(256) void vq_lse_rows(const float* __restrict__ cost,
                                                   const float* __restrict__ phi1,
                                                   float* __restrict__ acc) {
    int n = blockIdx.x, t = threadIdx.x, e = n >> 7;
    const float* row = cost + n * KCB;
    const float* ph  = phi1 + e * KCB;
    float m = -3.4e38f, s = 0.0f;
    for (int j = 0; j < 4; ++j) {
        int k = t + 256 * j;
        float x = (ph[k] - row[k]) * INV_EPS;
        if (x > m) { s = s * expf(m - x) + 1.0f; m = x; }
        else       { s += expf(x - m); }
    }
    __shared__ float sm[256]; __shared__ float ss[256];
    sm[t] = m; ss[t] = s; __syncthreads();
    for (int str = 128; str > 0; str >>= 1) {
        if (t < str) {
            float m1 = sm[t], s1 = ss[t], m2 = sm[t + str], s2 = ss[t + str];
            float mm = fmaxf(m1, m2);
            ss[t] = s1 * expf(m1 - mm) + s2 * expf(m2 - mm);
            sm[t] = mm;
        }
        __syncthreads();
    }
    if (t == 0) atomicAdd(&acc[0], sm[0] + logf(ss[0]));
}

// loss2: per (e,k) logsumexp over the 128 rows of ensemble e (log 8 folded into constants)
// grid: (e=64, ktile=16); 256 thr = 64 cols x 4 row-strips; k-major reads are coalesced
__global__ __launch_bounds__(256) void vq_lse_cols(const float* __restrict__ cost,
                                                   const float* __restrict__ phi2,
                                                   float* __restrict__ acc) {
    int e = blockIdx.x, k0 = blockIdx.y * 64;
    int t = threadIdx.x, tx = t & 63, ty = t >> 6;
    float m = -3.4e38f, s = 0.0f;
    for (int np = ty; np < 128; np += 4) {
        float x = (phi2[e * 128 + np] - cost[(e * 128 + np) * KCB + k0 + tx]) * INV_EPS;
        if (x > m) { s = s * expf(m - x) + 1.0f; m = x; }
        else       { s += expf(x - m); }
    }
    __shared__ float sm[256]; __shared__ float ss[256]; __shared__ float sl[64];
    sm[t] = m; ss[t] = s; __syncthreads();
    if (ty == 0) {
        float m1 = sm[tx], s1 = ss[tx];
        for (int j = 1; j < 4; ++j) {
            float m2 = sm[tx + 64 * j], s2 = ss[tx + 64 * j];
            float mm = fmaxf(m1, m2);
            s1 = s1 * expf(m1 - mm) + s2 * expf(m2 - mm);
            m1 = mm;
        }
        sl[tx] = m1 + logf(s1);
    }
    __syncthreads();
    for (int str = 32; str > 0; str >>= 1) {
        if (t < str) sl[t] += sl[t + str];
        __syncthreads();
    }
    if (t == 0) atomicAdd(&acc[1], sl[0]);
}

__global__ __launch_bounds__(256) void vq_finalize(const float* __restrict__ phi1,
                                                   const float* __restrict__ phi2,
                                                   const float* __restrict__ hist,
                                                   const float* __restrict__ acc,
                                                   float* __restrict__ out2) {
    int t = threadIdx.x;
    float s1 = 0.0f, s2 = 0.0f, ent = 0.0f;
    for (int i = t; i < ENS * KCB; i += 256) s1 += phi1[i];
    for (int i = t; i < N_TOK; i += 256)     s2 += phi2[i];
    for (int i = t; i < KCB; i += 256) {
        float em = hist[i] * (1.0f / 8192.0f);
        ent -= em * logf(em + 1e-10f);
    }
    __shared__ float r1[256]; __shared__ float r2[256]; __shared__ float r3[256];
    r1[t] = s1; r2[t] = s2; r3[t] = ent; __syncthreads();
    for (int s = 128; s > 0; s >>= 1) {
        if (t < s) { r1[t] += r1[t + s]; r2[t] += r2[t + s]; r3[t] += r3[t + s]; }
        __syncthreads();
    }
    if (t == 0) {
        float logInvK = logf(1.0f / 1024.0f);
        // loss1: rows tiled 8x -> weight (8/1024)=1/128 per distinct row
        float loss1 = (-TEMP_C) * (acc[0] * (1.0f / 128.0f) + 64.0f * logInvK)
                      + r1[0] * (1.0f / 1024.0f);
        // loss2: lse over tiled 1024 cols = log(8) + lse over 128 distinct
        float loss2 = (-TEMP_C) * (acc[1] * (1.0f / 1024.0f)
                                   + 64.0f * (logInvK + logf(8.0f)))
                      + r2[0] * (1.0f / 128.0f);
        out2[0] = BETA_C * (loss1 + loss2);
        out2[1] = expf(r3[0]);
    }
}

extern "C" void kernel_launch(void* const* d_in, const int* in_sizes, int n_in,
                              void* d_out, int out_size, void* d_ws, size_t ws_size,
                              hipStream_t stream) {
    const float* z   = (const float*)d_in[0];
    const float* cb  = (const float*)d_in[1];
    // d_in[2] = codebook_weight: softmax result unused downstream -> skipped
    const float* w1  = (const float*)d_in[3];
    const float* b1  = (const float*)d_in[4];
    const float* w2  = (const float*)d_in[5];
    const float* b2  = (const float*)d_in[6];

    float* ws   = (float*)d_ws;
    float* ZF   = ws + WS_ZF;
    float* COST = ws + WS_COST;
    float* ZN   = ws + WS_ZN;
    float* CK   = ws + WS_CK;
    float* PHI1 = ws + WS_PHI1;
    float* PHI2 = ws + WS_PHI2;
    float* HIST = ws + WS_HIST;
    int*   IDX  = (int*)(ws + WS_IDX);
    float* ACC  = ws + WS_ACC;

    float* out  = (float*)d_out;             // [2097152] zq, then loss, perplexity

    vq_init<<<1, 256, 0, stream>>>(HIST, ACC);
    vq_build_zf<<<N_TOK, 256, 0, stream>>>(z, ZF, ZN);
    vq_phi1_ck<<<KCB, 256, 0, stream>>>(cb, w1, b1, PHI1, CK);
    vq_phi2<<<N_TOK, 256, 0, stream>>>(ZF, w2, b2, PHI2);

    dim3 gGemm(N_TOK / 64, KCB / 32);
    vq_cost_wmma<<<gGemm, 256, 0, stream>>>(ZF, cb, ZN, CK, COST);

    vq_argmin<<<N_TOK, 256, 0, stream>>>(COST, IDX, HIST);
    vq_gather<<<(N_TOK * CDIM) / 256, 256, 0, stream>>>(cb, IDX, out);
    vq_lse_rows<<<N_TOK, 256, 0, stream>>>(COST, PHI1, ACC);
    dim3 gCols(ENS, KCB / 64);
    vq_lse_cols<<<gCols, 256, 0, stream>>>(COST, PHI2, ACC);
    vq_finalize<<<1, 256, 0, stream>>>(PHI1, PHI2, HIST, ACC, out + N_TOK * CDIM);
}